// HeteroGNN_49692771615165
// MI455X (gfx1250) — compile-verified
//
#include <hip/hip_runtime.h>

typedef __attribute__((ext_vector_type(2))) float v2f;
typedef __attribute__((ext_vector_type(8))) float v8f;

#define N_NODES 50000
#define N_EDGES 1600000
#define D_IN    128
#define D_HID   256

// ---------------- graph preprocessing ----------------

__global__ void zero_i32(int* __restrict__ p, int n) {
    int i = blockIdx.x * blockDim.x + threadIdx.x;
    if (i < n) p[i] = 0;
}

__global__ void count_deg(const long long* __restrict__ ei, int* __restrict__ cnt, int E) {
    int e = blockIdx.x * blockDim.x + threadIdx.x;
    if (e < E) atomicAdd(&cnt[(int)ei[(size_t)E + e]], 1);
}

// dinv = rsqrt(in_degree + 1 self loop); zero the CSR fill counters
__global__ void finalize_deg(const int* __restrict__ cnt, float* __restrict__ dinv,
                             int* __restrict__ fill, int n) {
    int i = blockIdx.x * blockDim.x + threadIdx.x;
    if (i < n) {
        dinv[i] = rsqrtf((float)(cnt[i] + 1));
        fill[i] = 0;
    }
}

// single-workgroup exclusive scan of cnt[0..n) -> rowptr[0..n]
__global__ __launch_bounds__(1024) void scan_rowptr(const int* __restrict__ cnt,
                                                    int* __restrict__ rowptr, int n) {
    __shared__ int sums[1024];
    int t = threadIdx.x;
    int chunk = (n + 1023) / 1024;
    int beg = t * chunk;
    int end = beg + chunk; if (end > n) end = n;
    int s = 0;
    for (int i = beg; i < end; ++i) s += cnt[i];
    sums[t] = s;
    __syncthreads();
    for (int off = 1; off < 1024; off <<= 1) {           // inclusive Hillis-Steele
        int v = (t >= off) ? sums[t - off] : 0;
        __syncthreads();
        sums[t] += v;
        __syncthreads();
    }
    int run = (t == 0) ? 0 : sums[t - 1];
    for (int i = beg; i < end; ++i) { rowptr[i] = run; run += cnt[i]; }
    if (end == n) rowptr[n] = run;                       // run == total here
}

__global__ void fill_csr(const long long* __restrict__ ei, const int* __restrict__ rowptr,
                         int* __restrict__ fill, int* __restrict__ csrc, int E) {
    int e = blockIdx.x * blockDim.x + threadIdx.x;
    if (e < E) {
        int s = (int)ei[e];
        int d = (int)ei[(size_t)E + e];
        int pos = rowptr[d] + atomicAdd(&fill[d], 1);
        csrc[pos] = s;
    }
}

// ---------------- WMMA fp32 GEMM: C[M,N] = A[M,K] @ B[K,N] ----------------
// Each wave computes a 16x64 strip: one A fragment (16x4 fp32) x 4 B fragments.
// 4 waves/block cover 4 consecutive M tiles. Tail guard is wave-uniform so
// EXEC is all-ones for every v_wmma (ISA requirement).

template <int K, int N>
__global__ __launch_bounds__(128) void gemm_wmma_f32(const float* __restrict__ A,
                                                     const float* __restrict__ B,
                                                     float* __restrict__ C, int M) {
    const int wave = threadIdx.x >> 5;
    const int lane = threadIdx.x & 31;
    const int tileM = (blockIdx.x * 4 + wave) * 16;
    const int n0 = blockIdx.y * 64;
    if (tileM >= M) return;                        // uniform within the wave

    const int l16   = lane & 15;
    const int khalf = (lane >> 4) << 1;            // lanes 0-15 -> K0/K1, 16-31 -> K2/K3
    const float* arow = A + (size_t)(tileM + l16) * K;

    v8f acc0 = {}, acc1 = {}, acc2 = {}, acc3 = {};
    for (int k0 = 0; k0 < K; k0 += 4) {
        const int kb = k0 + khalf;
        v2f a;
        a.x = arow[kb];
        a.y = arow[kb + 1];
        const float* bp = B + (size_t)kb * N + n0 + l16;
        v2f b0, b1, b2, b3;
        b0.x = bp[0];  b0.y = bp[N];
        b1.x = bp[16]; b1.y = bp[N + 16];
        b2.x = bp[32]; b2.y = bp[N + 32];
        b3.x = bp[48]; b3.y = bp[N + 48];
        acc0 = __builtin_amdgcn_wmma_f32_16x16x4_f32(false, a, false, b0, (short)0, acc0, false, false);
        acc1 = __builtin_amdgcn_wmma_f32_16x16x4_f32(false, a, false, b1, (short)0, acc1, false, false);
        acc2 = __builtin_amdgcn_wmma_f32_16x16x4_f32(false, a, false, b2, (short)0, acc2, false, false);
        acc3 = __builtin_amdgcn_wmma_f32_16x16x4_f32(false, a, false, b3, (short)0, acc3, false, false);
    }

    // C/D layout: VGPR i, lanes 0-15 -> row i, col=lane; lanes 16-31 -> row i+8
    const int rbase = tileM + (lane >> 4) * 8;
    const int cbase = n0 + l16;
    #pragma unroll
    for (int i = 0; i < 8; ++i) {
        float* crow = C + (size_t)(rbase + i) * N + cbase;
        crow[0]  = acc0[i];
        crow[16] = acc1[i];
        crow[32] = acc2[i];
        crow[48] = acc3[i];
    }
}

// ---------------- atomic-free normalized aggregation ----------------
// out[d] = act( dinv[d] * ( dinv[d]*h[d] + sum_in_edges dinv[s]*h[s] ) + bias )
// One block per node, one thread per feature column; coalesced row gathers.

template <int D, bool RELU>
__global__ __launch_bounds__(D) void aggregate(const float* __restrict__ h,
                                               const int* __restrict__ rowptr,
                                               const int* __restrict__ csrc,
                                               const float* __restrict__ dinv,
                                               const float* __restrict__ bias,
                                               float* __restrict__ out) {
    const int node = blockIdx.x;
    const int col  = threadIdx.x;
    const float di = dinv[node];
    float acc = di * h[(size_t)node * D + col];          // self loop (pre-scaled)
    const int e0 = rowptr[node], e1 = rowptr[node + 1];
    for (int e = e0; e < e1; ++e) {
        const int s = csrc[e];
        acc += dinv[s] * h[(size_t)s * D + col];
    }
    float v = di * acc + bias[col];
    if (RELU) v = fmaxf(v, 0.0f);
    out[(size_t)node * D + col] = v;
}

// ---------------- launch ----------------

extern "C" void kernel_launch(void* const* d_in, const int* in_sizes, int n_in,
                              void* d_out, int out_size, void* d_ws, size_t ws_size,
                              hipStream_t stream) {
    const float*     x  = (const float*)d_in[0];
    const long long* ei = (const long long*)d_in[1];   // [2, E] int64
    const float*     W1 = (const float*)d_in[2];
    const float*     b1 = (const float*)d_in[3];
    const float*     W2 = (const float*)d_in[4];
    const float*     b2 = (const float*)d_in[5];
    float* out = (float*)d_out;

    const int N = N_NODES, E = N_EDGES;

    char* ws = (char*)d_ws;
    size_t off = 0;
    auto carve = [&](size_t bytes) -> void* {
        void* p = ws + off;
        off += (bytes + 255) & ~(size_t)255;
        return p;
    };
    int*   cnt    = (int*)  carve((size_t)N * 4);
    int*   fill   = (int*)  carve((size_t)N * 4);
    int*   rowptr = (int*)  carve((size_t)(N + 1) * 4);
    float* dinv   = (float*)carve((size_t)N * 4);
    int*   csrc   = (int*)  carve((size_t)E * 4);
    float* bufA   = (float*)carve((size_t)N * D_HID * 4);  // h1, later reused for h2
    float* bufB   = (float*)carve((size_t)N * D_HID * 4);  // relu(agg1)

    // --- build CSR by destination (shared by both layers) ---
    zero_i32<<<(N + 255) / 256, 256, 0, stream>>>(cnt, N);
    count_deg<<<(E + 255) / 256, 256, 0, stream>>>(ei, cnt, E);
    finalize_deg<<<(N + 255) / 256, 256, 0, stream>>>(cnt, dinv, fill, N);
    scan_rowptr<<<1, 1024, 0, stream>>>(cnt, rowptr, N);
    fill_csr<<<(E + 255) / 256, 256, 0, stream>>>(ei, rowptr, fill, csrc, E);

    // --- layer 1: h1 = x @ W1 ; out1 = relu(agg(h1) + b1) ---
    {
        dim3 grid((N / 16 + 3) / 4, D_HID / 64);
        gemm_wmma_f32<D_IN, D_HID><<<grid, 128, 0, stream>>>(x, W1, bufA, N);
        aggregate<D_HID, true><<<N, D_HID, 0, stream>>>(bufA, rowptr, csrc, dinv, b1, bufB);
    }

    // --- layer 2: h2 = out1 @ W2 ; out = agg(h2) + b2 ---
    {
        dim3 grid((N / 16 + 3) / 4, D_IN / 64);
        gemm_wmma_f32<D_HID, D_IN><<<grid, 128, 0, stream>>>(bufB, W2, bufA, N);
        aggregate<D_IN, false><<<N, D_IN, 0, stream>>>(bufA, rowptr, csrc, dinv, b2, out);
    }
}